// SpectralConvLaplace3D_85727547228521
// MI455X (gfx1250) — compile-verified
//
#include <hip/hip_runtime.h>
#include <hip/hip_bf16.h>

// SpectralConvLaplace3D on MI455X (gfx1250).
// B=4, CI=CO=8, D=24 per axis, M1=M2=M3=4.
// DFT passes use V_WMMA_F32_16X16X4_F32 (fp32 matrix pipe, exact precision).

typedef __attribute__((ext_vector_type(2))) float v2f;
typedef __attribute__((ext_vector_type(8))) float v8f;

#define NB   4
#define NC   8          // CI == CO
#define ND   24
#define NM   4
#define SLAB (ND*ND*ND)         // 13824 complex per (b,c) slab
#define NSLAB (NB*NC)           // 32 slabs
#define NROWS (NSLAB*ND*ND)     // 18432 GEMM rows per DFT pass
#define CICO (NC*NC)            // 64

__device__ __forceinline__ float2 cmul(float2 a, float2 b) {
    return make_float2(a.x*b.x - a.y*b.y, a.x*b.y + a.y*b.x);
}
__device__ __forceinline__ float2 cadd(float2 a, float2 b) {
    return make_float2(a.x + b.x, a.y + b.y);
}

// ---------------------------------------------------------------------------
// Setup: A-tables Ai[f, cico, m] = 1/(lam_f - pole), t-tables Ti[cico, m, d] =
// exp(pole * d/23).  lam_f = 2*pi*i * fftfreq(24, 1/23) (purely imaginary).
// ---------------------------------------------------------------------------
__global__ void setup_tables_kernel(const float2* __restrict__ wp1,
                                    const float2* __restrict__ wp2,
                                    const float2* __restrict__ wp3,
                                    float2* __restrict__ A1, float2* __restrict__ A2,
                                    float2* __restrict__ A3,
                                    float2* __restrict__ T1, float2* __restrict__ T2,
                                    float2* __restrict__ T3) {
    const float TWO_PI = 6.28318530717958647692f;
    int t = threadIdx.x;
    // A tables: [f(24)][cico(64)][m(4)] = 6144 entries each
    for (int idx = t; idx < ND*CICO*NM; idx += 256) {
        int f    = idx >> 8;
        int cico = (idx >> 2) & 63;
        int m    = idx & 3;
        int fs   = (f >= ND/2) ? f - ND : f;
        float lamIm = TWO_PI * 23.0f * (float)fs / 24.0f;
        const float2* wps[3] = { wp1, wp2, wp3 };
        float2* As[3] = { A1, A2, A3 };
        for (int k = 0; k < 3; ++k) {
            float2 p = wps[k][cico*NM + m];
            float a = -p.x, b = lamIm - p.y;       // lam - pole
            float inv = 1.0f / (a*a + b*b);
            As[k][idx] = make_float2(a*inv, -b*inv); // 1/(a+bi)
        }
    }
    // t tables: [cico(64)][m(4)][d(24)] = 6144 entries each
    for (int idx = t; idx < CICO*NM*ND; idx += 256) {
        int cico = idx / (NM*ND);
        int m    = (idx / ND) & 3;
        int d    = idx % ND;
        float td = (float)d / 23.0f;
        const float2* wps[3] = { wp1, wp2, wp3 };
        float2* Ts[3] = { T1, T2, T3 };
        for (int k = 0; k < 3; ++k) {
            float2 p = wps[k][cico*NM + m];
            float e = expf(p.x * td);
            float s, c;
            sincosf(p.y * td, &s, &c);
            Ts[k][idx] = make_float2(e*c, e*s);
        }
    }
}

// ---------------------------------------------------------------------------
// G[fidx, ci, co] = sum_{m1,m2,m3} wr[cico,m]*A1[f1,m1]*A2[f2,m2]*A3[f3,m3]
// ---------------------------------------------------------------------------
__global__ void build_G_kernel(const float2* __restrict__ wr,
                               const float2* __restrict__ A1,
                               const float2* __restrict__ A2,
                               const float2* __restrict__ A3,
                               float2* __restrict__ G) {
    int gid  = blockIdx.x * 256 + threadIdx.x;   // < 884736
    int cico = gid & 63;
    int fidx = gid >> 6;
    int f3 = fidx % ND;
    int f2 = (fidx / ND) % ND;
    int f1 = fidx / (ND*ND);
    const float2* a1 = &A1[f1*(CICO*NM) + cico*NM];
    const float2* a2 = &A2[f2*(CICO*NM) + cico*NM];
    const float2* a3 = &A3[f3*(CICO*NM) + cico*NM];
    const float2* w  = &wr[cico*64];
    float2 acc = make_float2(0.f, 0.f);
    for (int m1 = 0; m1 < NM; ++m1) {
        float2 x1 = a1[m1];
        for (int m2 = 0; m2 < NM; ++m2) {
            float2 x12 = cmul(x1, a2[m2]);
            for (int m3 = 0; m3 < NM; ++m3)
                acc = cadd(acc, cmul(cmul(x12, a3[m3]), w[m1*16 + m2*4 + m3]));
        }
    }
    G[fidx*CICO + cico] = acc;
}

// ---------------------------------------------------------------------------
// DFT pass via FP32 WMMA. Batched complex GEMM: out[s, f, r] = sum_n
// in[s, r, n] * W[n, f],  W[n,f] = scale * exp(i*sign*2*pi*n*f/24).
// One wave per block; block computes a 16-row x 24-col complex tile.
// Twiddles stored in LDS as k-pairs (float2 {W[2k][f], W[2k+1][f]}) so each
// B fragment is a single aligned ds_load_b64 into a consecutive VGPR pair.
// A: 16x4 f32 fragment (VGPR0=K{0,2}, VGPR1=K{1,3} across lane halves);
// B mirrored; C/D per ISA 16x16 f32 layout.
// ---------------------------------------------------------------------------
template<bool REAL_IN>
__global__ void __launch_bounds__(32)
dft_pass_kernel(const float* __restrict__ inR, const float2* __restrict__ inC,
                float2* __restrict__ outC, float sign, float scale) {
    __shared__ float2 sWr2[12][32];   // {Wre[2k][f], Wre[2k+1][f]}
    __shared__ float2 sWi2[12][32];   // {Wim[2k][f], Wim[2k+1][f]}
    const int lane = threadIdx.x;
    for (int idx = lane; idx < 12*32; idx += 32) {
        int k2 = idx >> 5, f = idx & 31;
        float2 wre = make_float2(0.f, 0.f);
        float2 wim = make_float2(0.f, 0.f);
        if (f < 24) {
            const float step = 6.28318530717958647692f / 24.0f;
            float s0, c0, s1, c1;
            sincosf(step * (float)((2*k2)     * f), &s0, &c0);
            sincosf(step * (float)((2*k2 + 1) * f), &s1, &c1);
            wre = make_float2(scale * c0, scale * c1);
            wim = make_float2(scale * sign * s0, scale * sign * s1);
        }
        sWr2[k2][f] = wre;
        sWi2[k2][f] = wim;
    }
    __syncthreads();

    const int rb  = blockIdx.x * 16;     // 16-row tile; 576%16==0 -> one slab
    const int rlo = lane & 15;
    const int hi  = lane >> 4;           // 0: K={0,1}(+4k)   1: K={2,3}(+4k)
    const int m   = rb + rlo;            // A row fed by this lane

    v8f cr0 = {0,0,0,0,0,0,0,0}, ci0 = {0,0,0,0,0,0,0,0};
    v8f cr1 = {0,0,0,0,0,0,0,0}, ci1 = {0,0,0,0,0,0,0,0};

    for (int ks = 0; ks < 6; ++ks) {
        int kb = ks * 4 + hi * 2;        // always even
        int kp = kb >> 1;                // pair index into sW*2
        v2f ar, ai;
        if (REAL_IN) {
            ar = (v2f){ inR[m*24 + kb], inR[m*24 + kb + 1] };
            ai = (v2f){ 0.f, 0.f };
        } else {
            float2 c0 = inC[m*24 + kb];
            float2 c1 = inC[m*24 + kb + 1];
            ar = (v2f){ c0.x, c1.x };
            ai = (v2f){ c0.y, c1.y };
        }
        v2f nai = (v2f){ -ai.x, -ai.y };

        {   // N-tile 0: cols 0..15
            float2 r2 = sWr2[kp][rlo];
            float2 i2 = sWi2[kp][rlo];
            v2f br = (v2f){ r2.x, r2.y };
            v2f bi = (v2f){ i2.x, i2.y };
            cr0 = __builtin_amdgcn_wmma_f32_16x16x4_f32(false, ar, false, br, (short)0, cr0, false, false);
            ci0 = __builtin_amdgcn_wmma_f32_16x16x4_f32(false, ar, false, bi, (short)0, ci0, false, false);
            if (!REAL_IN) {
                cr0 = __builtin_amdgcn_wmma_f32_16x16x4_f32(false, nai, false, bi, (short)0, cr0, false, false);
                ci0 = __builtin_amdgcn_wmma_f32_16x16x4_f32(false, ai,  false, br, (short)0, ci0, false, false);
            }
        }
        {   // N-tile 1: cols 16..31 (24..31 multiply zero twiddles)
            float2 r2 = sWr2[kp][16 + rlo];
            float2 i2 = sWi2[kp][16 + rlo];
            v2f br = (v2f){ r2.x, r2.y };
            v2f bi = (v2f){ i2.x, i2.y };
            cr1 = __builtin_amdgcn_wmma_f32_16x16x4_f32(false, ar, false, br, (short)0, cr1, false, false);
            ci1 = __builtin_amdgcn_wmma_f32_16x16x4_f32(false, ar, false, bi, (short)0, ci1, false, false);
            if (!REAL_IN) {
                cr1 = __builtin_amdgcn_wmma_f32_16x16x4_f32(false, nai, false, bi, (short)0, cr1, false, false);
                ci1 = __builtin_amdgcn_wmma_f32_16x16x4_f32(false, ai,  false, br, (short)0, ci1, false, false);
            }
        }
    }

    // D layout: VGPR v -> M = hi*8+v, N = rlo. Transposed slab write.
    int s     = rb / 576;
    int rbase = (rb % 576) + hi * 8;
    for (int v = 0; v < 8; ++v) {
        int r  = rbase + v;
        int f0 = rlo;
        outC[s*SLAB + f0*576 + r] = make_float2(cr0[v], ci0[v]);
        int f1 = 16 + rlo;
        if (f1 < 24)
            outC[s*SLAB + f1*576 + r] = make_float2(cr1[v], ci1[v]);
    }
}

// ---------------------------------------------------------------------------
// or1[b, co, fidx] = sum_ci alpha[b, ci, fidx] * G[fidx, ci, co]
// ---------------------------------------------------------------------------
__global__ void mix_kernel(const float2* __restrict__ alpha,
                           const float2* __restrict__ G,
                           float2* __restrict__ or1) {
    int gid  = blockIdx.x * 256 + threadIdx.x;   // < 442368
    int fidx = gid % SLAB;
    int bco  = gid / SLAB;
    int b = bco >> 3, co = bco & 7;
    float2 acc = make_float2(0.f, 0.f);
    for (int ci = 0; ci < NC; ++ci)
        acc = cadd(acc, cmul(alpha[(b*NC + ci)*SLAB + fidx], G[fidx*CICO + ci*NC + co]));
    or1[gid] = acc;
}

// ---------------------------------------------------------------------------
// or2 chain: U -> V -> W -> or2 (staged contractions over f3, f2, f1, ci)
// ---------------------------------------------------------------------------
__global__ void u_kernel(const float2* __restrict__ alpha,
                         const float2* __restrict__ A3, float2* __restrict__ U) {
    int gid  = blockIdx.x * 256 + threadIdx.x;  // < 589824
    int f12  = gid % 576;
    int rest = gid / 576;                       // b*256 + ci*32 + co*4 + m3
    int m3   = rest & 3;
    int cico = (rest >> 2) & 63;
    int bci  = rest >> 5;
    const float2* al = &alpha[bci*SLAB + f12*24];
    float2 acc = make_float2(0.f, 0.f);
    for (int f3 = 0; f3 < ND; ++f3)
        acc = cadd(acc, cmul(al[f3], A3[f3*(CICO*NM) + cico*NM + m3]));
    U[gid] = acc;
}

__global__ void v_kernel(const float2* __restrict__ U,
                         const float2* __restrict__ A2, float2* __restrict__ V) {
    int gid   = blockIdx.x * 256 + threadIdx.x; // < 98304
    int f1    = gid % 24;
    int rest  = gid / 24;                       // bcico*16 + m2*4 + m3
    int m3    = rest & 3;
    int m2    = (rest >> 2) & 3;
    int bcico = rest >> 4;
    int cico  = bcico & 63;
    float2 acc = make_float2(0.f, 0.f);
    for (int f2 = 0; f2 < ND; ++f2)
        acc = cadd(acc, cmul(U[(bcico*NM + m3)*576 + f1*24 + f2],
                             A2[f2*(CICO*NM) + cico*NM + m2]));
    V[gid] = acc;
}

__global__ void w_kernel(const float2* __restrict__ V,
                         const float2* __restrict__ A1, float2* __restrict__ Wt) {
    int gid   = blockIdx.x * 256 + threadIdx.x; // < 16384
    int m3    = gid & 3;
    int m2    = (gid >> 2) & 3;
    int m1    = (gid >> 4) & 3;
    int bcico = gid >> 6;
    int cico  = bcico & 63;
    float2 acc = make_float2(0.f, 0.f);
    for (int f1 = 0; f1 < ND; ++f1)
        acc = cadd(acc, cmul(V[(bcico*16 + m2*4 + m3)*24 + f1],
                             A1[f1*(CICO*NM) + cico*NM + m1]));
    Wt[gid] = acc;
}

__global__ void or2_kernel(const float2* __restrict__ Wt,
                           const float2* __restrict__ wr, float2* __restrict__ or2) {
    int gid = blockIdx.x * 256 + threadIdx.x;   // < 2048
    int mmm = gid & 63;
    int bco = gid >> 6;
    int b = bco >> 3, co = bco & 7;
    float2 acc = make_float2(0.f, 0.f);
    for (int ci = 0; ci < NC; ++ci) {
        int cico = ci*NC + co;
        acc = cadd(acc, cmul(wr[cico*64 + mmm], Wt[(b*64 + cico)*64 + mmm]));
    }
    or2[gid] = make_float2(-acc.x, -acc.y);     // einsum uses -Hw
}

// ---------------------------------------------------------------------------
// x2 chain: P (m1 contraction) -> Q (m2) -> final (c,m3 + add Re(ifft))
// ---------------------------------------------------------------------------
__global__ void p_kernel(const float2* __restrict__ or2,
                         const float2* __restrict__ T1, float2* __restrict__ P) {
    int gid  = blockIdx.x * 256 + threadIdx.x;  // < 98304
    int z    = gid % 24;
    int rest = gid / 24;                        // bcj*16 + m2*4 + m3
    int m3   = rest & 3;
    int m2   = (rest >> 2) & 3;
    int bcj  = rest >> 4;
    int j    = bcj & 7;
    int bc   = bcj >> 3;
    int c    = bc & 7;
    int b    = bc >> 3;
    int cj   = c*NC + j;
    float2 acc = make_float2(0.f, 0.f);
    for (int m1 = 0; m1 < NM; ++m1)
        acc = cadd(acc, cmul(or2[(b*NC + c)*64 + m1*16 + m2*4 + m3],
                             T1[cj*(NM*ND) + m1*ND + z]));
    P[gid] = acc;
}

__global__ void q_kernel(const float2* __restrict__ P,
                         const float2* __restrict__ T2, float2* __restrict__ Q) {
    int gid  = blockIdx.x * 256 + threadIdx.x;  // < 589824
    int zx   = gid % 576;
    int z    = zx / 24, ix = zx % 24;
    int rest = gid / 576;                       // bcj*4 + m3
    int m3   = rest & 3;
    int bcj  = rest >> 2;
    int j    = bcj & 7;
    int bc   = bcj >> 3;
    int c    = bc & 7;
    int cj   = c*NC + j;
    float2 acc = make_float2(0.f, 0.f);
    for (int m2 = 0; m2 < NM; ++m2)
        acc = cadd(acc, cmul(P[(bcj*16 + m2*4 + m3)*24 + z],
                             T2[cj*(NM*ND) + m2*ND + ix]));
    Q[gid] = acc;
}

__global__ void final_kernel(const float2* __restrict__ Q,
                             const float2* __restrict__ T3,
                             const float2* __restrict__ ifftRes,
                             float* __restrict__ out) {
    int gid  = blockIdx.x * 256 + threadIdx.x;  // < 442368
    int sidx = gid % SLAB;
    int bj   = gid / SLAB;
    int b = bj >> 3, j = bj & 7;
    int z   = sidx / 576;
    int rem = sidx % 576;
    int d2  = rem / 24, d3 = rem % 24;
    float acc = 0.f;                             // real part only
    for (int c = 0; c < NC; ++c) {
        int cj  = c*NC + j;
        int bcj = (b*NC + c)*NC + j;
        for (int m3 = 0; m3 < NM; ++m3) {
            float2 q = Q[(bcj*NM + m3)*576 + z*24 + d2];
            float2 t = T3[cj*(NM*ND) + m3*ND + d3];
            acc += q.x*t.x - q.y*t.y;
        }
    }
    out[gid] = ifftRes[bj*SLAB + sidx].x + acc * (1.0f / 13824.0f);
}

// ---------------------------------------------------------------------------
// Host side: workspace layout (float2 units) and launch sequence.
// ---------------------------------------------------------------------------
extern "C" void kernel_launch(void* const* d_in, const int* in_sizes, int n_in,
                              void* d_out, int out_size, void* d_ws, size_t ws_size,
                              hipStream_t stream) {
    const float*  x   = (const float*)d_in[0];
    const float2* wp1 = (const float2*)d_in[1];
    const float2* wp2 = (const float2*)d_in[2];
    const float2* wp3 = (const float2*)d_in[3];
    const float2* wr  = (const float2*)d_in[4];
    float* out = (float*)d_out;

    float2* ws = (float2*)d_ws;
    size_t off = 0;
    float2* A1   = ws + off; off += ND*CICO*NM;      // 6144
    float2* A2   = ws + off; off += ND*CICO*NM;
    float2* A3   = ws + off; off += ND*CICO*NM;
    float2* T1   = ws + off; off += CICO*NM*ND;
    float2* T2   = ws + off; off += CICO*NM*ND;
    float2* T3   = ws + off; off += CICO*NM*ND;
    float2* G    = ws + off; off += (size_t)SLAB*CICO;   // 884736
    float2* bufA = ws + off; off += (size_t)NSLAB*SLAB;  // 442368
    float2* bufB = ws + off; off += (size_t)NSLAB*SLAB;
    float2* alph = ws + off; off += (size_t)NSLAB*SLAB;
    float2* or1  = ws + off; off += (size_t)NSLAB*SLAB;
    float2* ifr  = ws + off; off += (size_t)NSLAB*SLAB;
    float2* U    = ws + off; off += 589824;
    float2* V    = ws + off; off += 98304;
    float2* Wt   = ws + off; off += 16384;
    float2* or2v = ws + off; off += 2048;
    float2* P    = ws + off; off += 98304;
    float2* Q    = ws + off; off += 589824;
    (void)ws_size; (void)in_sizes; (void)n_in; (void)out_size;

    const int GEMM_BLOCKS = NROWS / 16;  // 1152

    setup_tables_kernel<<<1, 256, 0, stream>>>(wp1, wp2, wp3, A1, A2, A3, T1, T2, T3);
    build_G_kernel<<<(SLAB*CICO)/256, 256, 0, stream>>>(wr, A1, A2, A3, G);

    // forward FFT: x -> bufA -> bufB -> alpha   (sign=-1, scale=1)
    dft_pass_kernel<true ><<<GEMM_BLOCKS, 32, 0, stream>>>(x, nullptr, bufA, -1.0f, 1.0f);
    dft_pass_kernel<false><<<GEMM_BLOCKS, 32, 0, stream>>>(nullptr, bufA, bufB, -1.0f, 1.0f);
    dft_pass_kernel<false><<<GEMM_BLOCKS, 32, 0, stream>>>(nullptr, bufB, alph, -1.0f, 1.0f);

    mix_kernel<<<(NSLAB*SLAB)/256, 256, 0, stream>>>(alph, G, or1);

    u_kernel  <<<589824/256, 256, 0, stream>>>(alph, A3, U);
    v_kernel  <<< 98304/256, 256, 0, stream>>>(U, A2, V);
    w_kernel  <<< 16384/256, 256, 0, stream>>>(V, A1, Wt);
    or2_kernel<<<  2048/256, 256, 0, stream>>>(Wt, wr, or2v);

    // inverse FFT: or1 -> bufA -> bufB -> ifr   (sign=+1, scale=1/24 per pass)
    dft_pass_kernel<false><<<GEMM_BLOCKS, 32, 0, stream>>>(nullptr, or1,  bufA, 1.0f, 1.0f/24.0f);
    dft_pass_kernel<false><<<GEMM_BLOCKS, 32, 0, stream>>>(nullptr, bufA, bufB, 1.0f, 1.0f/24.0f);
    dft_pass_kernel<false><<<GEMM_BLOCKS, 32, 0, stream>>>(nullptr, bufB, ifr,  1.0f, 1.0f/24.0f);

    p_kernel<<< 98304/256, 256, 0, stream>>>(or2v, T1, P);
    q_kernel<<<589824/256, 256, 0, stream>>>(P, T2, Q);
    final_kernel<<<(NSLAB*SLAB)/256, 256, 0, stream>>>(Q, T3, ifr, out);
}